// EnhancedQuantumLayer_89292370083847
// MI455X (gfx1250) — compile-verified
//
#include <hip/hip_runtime.h>
#include <hip/hip_bf16.h>

typedef float v2f __attribute__((ext_vector_type(2)));
typedef float v8f __attribute__((ext_vector_type(8)));

#define NQ 6
#define DIM 64          // 2^NQ
#define NL 6

// ---------------------------------------------------------------------------
// Kernel 1: build the shared 64x64 unitary M (Rot layers + CNOT rings) by
// applying the gate sequence to the 64 basis columns. One workgroup, 64
// threads, thread j produces column j. Output layout in d_ws:
//   Mr at M[n*64 + j], Mi at M[4096 + n*64 + j]   (row n, col j)
// Performance irrelevant (runs once per launch, 2 waves).
// ---------------------------------------------------------------------------
__global__ void build_unitary_kernel(const float* __restrict__ w,
                                     float* __restrict__ M) {
    const int j = threadIdx.x;            // column index 0..63
    float sr[DIM], si[DIM];
    for (int k = 0; k < DIM; ++k) { sr[k] = (k == j) ? 1.f : 0.f; si[k] = 0.f; }

    for (int l = 0; l < NL; ++l) {
        // Rot(phi,theta,omega) on each qubit
        for (int q = 0; q < NQ; ++q) {
            const float* g = w + (l * NQ + q) * 3;
            float phi = g[0], th = g[1], om = g[2];
            float stt, ct, shs, chs, shd, chd;
            __sincosf(0.5f * th, &stt, &ct);
            __sincosf(0.5f * (phi + om), &shs, &chs);
            __sincosf(0.5f * (phi - om), &shd, &chd);
            // U00 = ct*e^{-i(phi+om)/2}; U01 = -st*e^{+i(phi-om)/2}
            // U10 = st*e^{-i(phi-om)/2}; U11 = ct*e^{+i(phi+om)/2}
            float u00r =  ct  * chs, u00i = -ct  * shs;
            float u01r = -stt * chd, u01i = -stt * shd;
            float u10r =  stt * chd, u10i = -stt * shd;
            float u11r =  ct  * chs, u11i =  ct  * shs;
            int mask = 1 << (5 - q);      // qubit q lives at bit (NQ-1-q)
            for (int k = 0; k < DIM; ++k) {
                if (k & mask) continue;
                int k1 = k | mask;
                float ar = sr[k],  ai = si[k];
                float br = sr[k1], bi = si[k1];
                sr[k]  = u00r*ar - u00i*ai + u01r*br - u01i*bi;
                si[k]  = u00r*ai + u00i*ar + u01r*bi + u01i*br;
                sr[k1] = u10r*ar - u10i*ai + u11r*br - u11i*bi;
                si[k1] = u10r*ai + u10i*ar + u11r*bi + u11i*br;
            }
        }
        // entangling ring: CNOT(q, (q+r)%NQ), r = (l % (NQ-1)) + 1
        int r = (l % (NQ - 1)) + 1;
        for (int q = 0; q < NQ; ++q) {
            int t  = (q + r) % NQ;
            int cm = 1 << (5 - q), tm = 1 << (5 - t);
            for (int k = 0; k < DIM; ++k) {
                if ((k & cm) && !(k & tm)) {
                    int k1 = k | tm;
                    float tr = sr[k], ti = si[k];
                    sr[k] = sr[k1];  si[k] = si[k1];
                    sr[k1] = tr;     si[k1] = ti;
                }
            }
        }
    }
    for (int n = 0; n < DIM; ++n) {
        M[n * DIM + j]        = sr[n];   // Mr[n][j]
        M[4096 + n * DIM + j] = si[n];   // Mi[n][j]
    }
}

// ---------------------------------------------------------------------------
// Kernel 2: fused embed + GEMM (WMMA f32 16x16x4) + PauliZ epilogue.
// Each wave computes a 16-sample x 128-packed-output strip:
//   out_packed[b, 2n]   = Re psi_n[b],  out_packed[b, 2n+1] = Im psi_n[b]
//   A[b, 2k] = Re v_k[b], A[b, 2k+1] = Im v_k[b]  (embedding product state,
//   generated analytically per lane, never touches memory)
//   B[2k,2n]=Mr[n,k]  B[2k+1,2n]=-Mi[n,k]  B[2k,2n+1]=Mi[n,k]  B[2k+1,2n+1]=Mr[n,k]
// 256 threads (8 waves), 128 samples per workgroup.
// ---------------------------------------------------------------------------
__global__ __launch_bounds__(256)
void quantum_wmma_kernel(const float* __restrict__ x,
                         const float* __restrict__ scale_p,
                         const float* __restrict__ Mg,
                         float* __restrict__ out) {
    // LDS: unitary stored transposed so GEMM reads sMr[k*64+n] are
    // conflict-free (lanes differ in n).
    __shared__ float sMr[4096];
    __shared__ float sMi[4096];
    const int tid = threadIdx.x;
    for (int i = tid; i < 4096; i += 256) {
        int k = i >> 6, n = i & 63;
        sMr[i] = Mg[n * DIM + k];
        sMi[i] = Mg[4096 + n * DIM + k];
    }
    __syncthreads();

    const int lane = tid & 31;
    const int wave = tid >> 5;          // 0..7
    const int c    = lane & 15;         // position within 16-lane half
    const int hi   = lane >> 4;         // which half
    const int strip = blockIdx.x * 128 + wave * 16;
    const int srow  = strip + c;        // this lane's sample row for A frags

    // embedding trig for this lane's sample: cq/sq per qubit
    const float s = scale_p[0];
    float cq[NQ], sq[NQ];
#pragma unroll
    for (int q = 0; q < NQ; ++q) {
        float ang = 0.5f * s * x[srow * NQ + q];
        __sincosf(ang, &sq[q], &cq[q]);
    }

    v8f acc[8];
#pragma unroll
    for (int t = 0; t < 8; ++t)
#pragma unroll
        for (int e = 0; e < 8; ++e) acc[t][e] = 0.f;

    // K loop over packed K=128 in chunks of 4 (one basis index per lane/iter)
    for (int k0 = 0; k0 < 128; k0 += 4) {
        const int k = (k0 >> 1) + hi;   // basis state index 0..63
        // v[k] = (-i)^popcount(k) * prod_q (bit ? sin : cos)
        float mag = 1.f;
#pragma unroll
        for (int q = 0; q < NQ; ++q)
            mag *= ((k >> (5 - q)) & 1) ? sq[q] : cq[q];
        const int pc = __popc(k) & 3;
        const float vr = (pc == 0) ? mag : ((pc == 2) ? -mag : 0.f);
        const float vi = (pc == 1) ? -mag : ((pc == 3) ? mag : 0.f);
        v2f a; a.x = vr; a.y = vi;      // A frag: K = k0+2*hi (+1)

#pragma unroll
        for (int t = 0; t < 8; ++t) {
            const int nprime = t * 16 + c;       // packed output column
            const int n = nprime >> 1;
            const float mr = sMr[k * DIM + n];
            const float mi = sMi[k * DIM + n];
            v2f b;
            if (nprime & 1) { b.x = mi; b.y = mr; }   // imag output column
            else            { b.x = mr; b.y = -mi; }  // real output column
            acc[t] = __builtin_amdgcn_wmma_f32_16x16x4_f32(
                false, a, false, b, (short)0, acc[t], false, false);
        }
    }

    // Epilogue: |psi|^2 and the six <Z_q> sums, all via wave shuffles.
    // D layout: VGPR r -> sample row (r + 8*hi), column n' = t*16 + c.
#pragma unroll
    for (int r = 0; r < 8; ++r) {
        float ev0 = 0.f, ev1 = 0.f, ev2 = 0.f, ev3 = 0.f, ev4 = 0.f, ev5 = 0.f;
#pragma unroll
        for (int t = 0; t < 8; ++t) {
            float val = acc[t][r];
            float nb  = __shfl_xor(val, 1, 32);          // re/im partner
            float p   = ((c & 1) == 0) ? (val * val + nb * nb) : 0.f;
            int jdx   = t * 8 + (c >> 1);                // basis index
            ev0 += ((jdx >> 5) & 1) ? -p : p;            // qubit 0 = bit 5
            ev1 += ((jdx >> 4) & 1) ? -p : p;
            ev2 += ((jdx >> 3) & 1) ? -p : p;
            ev3 += ((jdx >> 2) & 1) ? -p : p;
            ev4 += ((jdx >> 1) & 1) ? -p : p;
            ev5 += ( jdx       & 1) ? -p : p;
        }
        // reduce over the 16-lane half (masks < 16 never cross halves)
#pragma unroll
        for (int m = 1; m <= 8; m <<= 1) {
            ev0 += __shfl_xor(ev0, m, 32);
            ev1 += __shfl_xor(ev1, m, 32);
            ev2 += __shfl_xor(ev2, m, 32);
            ev3 += __shfl_xor(ev3, m, 32);
            ev4 += __shfl_xor(ev4, m, 32);
            ev5 += __shfl_xor(ev5, m, 32);
        }
        const int sample = strip + r + 8 * hi;
        float evsel = (c == 0) ? ev0 : (c == 1) ? ev1 : (c == 2) ? ev2
                    : (c == 3) ? ev3 : (c == 4) ? ev4 : ev5;
        if (c < NQ) out[sample * NQ + c] = evsel;
    }
}

extern "C" void kernel_launch(void* const* d_in, const int* in_sizes, int n_in,
                              void* d_out, int out_size, void* d_ws, size_t ws_size,
                              hipStream_t stream) {
    (void)n_in; (void)out_size; (void)ws_size;
    const float* x     = (const float*)d_in[0];   // (B, 6) f32
    const float* w     = (const float*)d_in[1];   // (6, 6, 3) f32
    const float* scale = (const float*)d_in[2];   // (1,) f32
    float* M   = (float*)d_ws;                    // 32 KB: Mr | Mi
    float* out = (float*)d_out;                   // (B, 6) f32

    const int B = in_sizes[0] / NQ;               // 32768
    const int grid = B / 128;                     // 128 samples per WG

    build_unitary_kernel<<<1, 64, 0, stream>>>(w, M);
    quantum_wmma_kernel<<<grid, 256, 0, stream>>>(x, scale, M, out);
}